// GravNetBlock_31044023615652
// MI455X (gfx1250) — compile-verified
//
#include <hip/hip_runtime.h>
#include <math.h>

typedef __attribute__((ext_vector_type(2))) float v2f;
typedef __attribute__((ext_vector_type(8))) float v8f;

#define NTOT 16384   // B*N
#define NB   8192    // nodes per batch
#define CIN  9
#define DD   32
#define SS   3
#define KNN  40
#define BN_EPS 1e-5f

// ---------------------------------------------------------------- WMMA helpers
__device__ __forceinline__ v8f wmma4(v2f a, v2f b, v8f c) {
  // V_WMMA_F32_16X16X4_F32 : D(16x16 f32) = A(16x4 f32) * B(4x16 f32) + C
  return __builtin_amdgcn_wmma_f32_16x16x4_f32(false, a, false, b, (short)0, c,
                                               false, false);
}

// A 16x4 fragment from global row-major X (no padding needed: k0+3 < ld)
__device__ __forceinline__ v2f load_a(const float* __restrict__ X, int ld,
                                      int row0, int k0, int lane) {
  int m = lane & 15;
  int k = k0 + ((lane & 16) ? 2 : 0);
  const float* p = X + (size_t)(row0 + m) * ld;
  v2f a;
  a.x = p[k];
  a.y = p[k + 1];
  return a;
}

// B 4x16 fragment from full (unpadded) row-major W[K][32], col tile n0
__device__ __forceinline__ v2f load_b(const float* __restrict__ W, int n0,
                                      int k0, int lane) {
  int n = (lane & 15) + n0;
  int k = k0 + ((lane & 16) ? 2 : 0);
  v2f b;
  b.x = W[(size_t)k * DD + n];
  b.y = W[(size_t)(k + 1) * DD + n];
  return b;
}

// D 16x16 store: lane L: col n=(L&15)+n0; vgpr r -> row row0 + r + (L&16?8:0)
__device__ __forceinline__ void store_d(float* __restrict__ O, int ld, int row0,
                                        int n0, int lane, v8f d) {
  int n = (lane & 15) + n0;
  int rbase = row0 + ((lane & 16) ? 8 : 0);
#pragma unroll
  for (int r = 0; r < 8; ++r) O[(size_t)(rbase + r) * ld + n] = d[r];
}

__device__ __forceinline__ v8f elu8(v8f x) {
  v8f r;
#pragma unroll
  for (int i = 0; i < 8; ++i) r[i] = x[i] > 0.0f ? x[i] : __expf(x[i]) - 1.0f;
  return r;
}

// ---------------------------------------------------------------- kernels
__global__ void k_init(float* bnstats) {
  if (threadIdx.x < 128) bnstats[threadIdx.x] = 0.0f;
}

// pre_gravnet: ELU(ELU(x@W1+b1)@W2+b2) -> x1_raw, accumulate BN1 sum/sumsq
__global__ void k_pre_mlp(const float* __restrict__ x,
                          const float* __restrict__ W1,
                          const float* __restrict__ b1,
                          const float* __restrict__ W2,
                          const float* __restrict__ b2,
                          float* __restrict__ x1_raw,
                          float* __restrict__ bnstats) {
  __shared__ float w1p[12][32];     // W1 zero-padded K 9->12
  __shared__ float xa[8][16][12];   // per-wave x tile zero-padded
  __shared__ float xp[8][16][32];   // layer-1 output transpose staging
  __shared__ float bnacc[64];
  int tid = threadIdx.x;
  if (tid < 64) bnacc[tid] = 0.0f;
  for (int i = tid; i < 12 * 32; i += 256) {
    int kr = i >> 5, c = i & 31;
    w1p[kr][c] = (kr < CIN) ? W1[kr * DD + c] : 0.0f;
  }
  __syncthreads();
  int lane = tid & 31, w = tid >> 5;
  int row0 = blockIdx.x * 128 + w * 16;
  int n = lane & 15;

  // stage x tile (16 rows x 9 cols) zero-padded to 12 cols into LDS
  for (int i = lane; i < 16 * 12; i += 32) {
    int m = i / 12, k = i % 12;
    xa[w][m][k] = (k < CIN) ? x[(size_t)(row0 + m) * CIN + k] : 0.0f;
  }
  __builtin_amdgcn_wave_barrier();

  v8f a0 = {}, a1 = {};
#pragma unroll
  for (int k0 = 0; k0 < 12; k0 += 4) {
    int kk = k0 + ((lane & 16) ? 2 : 0);
    v2f av; av.x = xa[w][n][kk]; av.y = xa[w][n][kk + 1];
    v2f bv0; bv0.x = w1p[kk][n];      bv0.y = w1p[kk + 1][n];
    v2f bv1; bv1.x = w1p[kk][n + 16]; bv1.y = w1p[kk + 1][n + 16];
    a0 = wmma4(av, bv0, a0);
    a1 = wmma4(av, bv1, a1);
  }
  float bb0 = b1[n], bb1 = b1[n + 16];
#pragma unroll
  for (int r = 0; r < 8; ++r) { a0[r] += bb0; a1[r] += bb1; }
  a0 = elu8(a0); a1 = elu8(a1);

  // D-layout -> A-layout transpose through per-wave LDS tile (in-order per wave)
  int rb = (lane & 16) ? 8 : 0;
#pragma unroll
  for (int r = 0; r < 8; ++r) {
    xp[w][rb + r][n] = a0[r];
    xp[w][rb + r][n + 16] = a1[r];
  }
  __builtin_amdgcn_wave_barrier();

  v8f o0 = {}, o1 = {};
#pragma unroll
  for (int k0 = 0; k0 < 32; k0 += 4) {
    int kk = k0 + ((lane & 16) ? 2 : 0);
    v2f av; av.x = xp[w][n][kk]; av.y = xp[w][n][kk + 1];
    o0 = wmma4(av, load_b(W2, 0, k0, lane), o0);
    o1 = wmma4(av, load_b(W2, 16, k0, lane), o1);
  }
  bb0 = b2[n]; bb1 = b2[n + 16];
#pragma unroll
  for (int r = 0; r < 8; ++r) { o0[r] += bb0; o1[r] += bb1; }
  o0 = elu8(o0); o1 = elu8(o1);
  store_d(x1_raw, DD, row0, 0, lane, o0);
  store_d(x1_raw, DD, row0, 16, lane, o1);

  float s0 = 0, q0 = 0, s1 = 0, q1 = 0;
#pragma unroll
  for (int r = 0; r < 8; ++r) {
    s0 += o0[r]; q0 += o0[r] * o0[r];
    s1 += o1[r]; q1 += o1[r] * o1[r];
  }
  atomicAdd(&bnacc[n], s0);
  atomicAdd(&bnacc[n + 16], s1);
  atomicAdd(&bnacc[32 + n], q0);
  atomicAdd(&bnacc[32 + n + 16], q1);
  __syncthreads();
  if (tid < 64) atomicAdd(&bnstats[tid], bnacc[tid]);
}

// apply BN1, project h = x1@Wh+bh, s = x1@Ws; scatter x1/s into concat buffers
__global__ void k_bn1_proj(const float* __restrict__ x1_raw,
                           const float* __restrict__ bnstats,
                           const float* __restrict__ g,
                           const float* __restrict__ bvec,
                           const float* __restrict__ Wh,
                           const float* __restrict__ bh,
                           const float* __restrict__ Ws,
                           float* __restrict__ hbuf, float* __restrict__ sbuf,
                           float* __restrict__ cat96, float* __restrict__ xc) {
  __shared__ float sc[32], sh[32];
  __shared__ float wsp[32][16];     // Ws zero-padded N 3->16
  __shared__ float xp[8][16][32];
  int tid = threadIdx.x;
  if (tid < 32) {
    float mu = bnstats[tid] * (1.0f / NTOT);
    float var = bnstats[32 + tid] * (1.0f / NTOT) - mu * mu;
    float s = g[tid] * rsqrtf(var + BN_EPS);
    sc[tid] = s;
    sh[tid] = bvec[tid] - mu * s;
  }
  for (int i = tid; i < 32 * 16; i += 256) {
    int kr = i >> 4, c = i & 15;
    wsp[kr][c] = (c < SS) ? Ws[kr * SS + c] : 0.0f;
  }
  __syncthreads();
  int lane = tid & 31, w = tid >> 5;
  int row0 = blockIdx.x * 128 + w * 16;
  int n = lane & 15;

  // normalized x1: stage into LDS (A operand) + scatter into cat96/xc
  float scl = sc[lane], shl = sh[lane];
#pragma unroll
  for (int m = 0; m < 16; ++m) {
    int row = row0 + m;
    float v = x1_raw[(size_t)row * DD + lane] * scl + shl;
    xp[w][m][lane] = v;
    cat96[(size_t)row * 96 + 64 + lane] = v;
    xc[(size_t)row * 68 + 35 + lane] = v;
    if (lane == 0) xc[(size_t)row * 68 + 67] = 0.0f;  // pad col for K=68
  }
  __builtin_amdgcn_wave_barrier();

  v8f h0 = {}, h1 = {}, s0 = {};
#pragma unroll
  for (int k0 = 0; k0 < 32; k0 += 4) {
    int kk = k0 + ((lane & 16) ? 2 : 0);
    v2f av; av.x = xp[w][n][kk]; av.y = xp[w][n][kk + 1];
    v2f bs; bs.x = wsp[kk][n]; bs.y = wsp[kk + 1][n];
    h0 = wmma4(av, load_b(Wh, 0, k0, lane), h0);
    h1 = wmma4(av, load_b(Wh, 16, k0, lane), h1);
    s0 = wmma4(av, bs, s0);
  }
  float bb0 = bh[n], bb1 = bh[n + 16];
#pragma unroll
  for (int r = 0; r < 8; ++r) { h0[r] += bb0; h1[r] += bb1; }
  store_d(hbuf, DD, row0, 0, lane, h0);
  store_d(hbuf, DD, row0, 16, lane, h1);

  int rb = (lane & 16) ? 8 : 0;
  if (n < SS) {
#pragma unroll
    for (int r = 0; r < 8; ++r) {
      int row = row0 + rb + r;
      sbuf[(size_t)row * SS + n] = s0[r];
      xc[(size_t)row * 68 + 32 + n] = s0[r];
    }
  }
}

// brute-force kNN (K=40) per batch; distance keys in VGPRs, indices in LDS
__global__ void k_knn(const float* __restrict__ sbuf,
                      float* __restrict__ potbuf, int* __restrict__ idxbuf) {
  __shared__ float stile[256 * 3];
  __shared__ int idxs[128][KNN];
  int tid = threadIdx.x;
  int b = blockIdx.y;
  int node = blockIdx.x * 128 + tid;  // 0..NB-1
  const float* sb = sbuf + (size_t)b * NB * SS;
  float mx = sb[(size_t)node * 3 + 0];
  float my = sb[(size_t)node * 3 + 1];
  float mz = sb[(size_t)node * 3 + 2];

  float bestd[KNN];
#pragma unroll
  for (int k = 0; k < KNN; ++k) bestd[k] = INFINITY;
  float wmax = INFINITY;
  int wslot = 0;

  for (int t0 = 0; t0 < NB; t0 += 256) {
    __syncthreads();
    for (int i = tid; i < 256 * 3; i += 128) stile[i] = sb[(size_t)t0 * 3 + i];
    __syncthreads();
    for (int j = 0; j < 256; ++j) {
      int src = t0 + j;
      float dx = mx - stile[j * 3 + 0];
      float dy = my - stile[j * 3 + 1];
      float dz = mz - stile[j * 3 + 2];
      float d2 = fmaf(dx, dx, fmaf(dy, dy, dz * dz));
      if (src != node && d2 < wmax) {
        int slot = wslot;
        idxs[tid][slot] = src;  // dynamic index -> single LDS store
#pragma unroll
        for (int k = 0; k < KNN; ++k)
          if (k == slot) bestd[k] = d2;
        wmax = bestd[0]; wslot = 0;
#pragma unroll
        for (int k = 1; k < KNN; ++k)
          if (bestd[k] > wmax) { wmax = bestd[k]; wslot = k; }
      }
    }
  }
  size_t base = ((size_t)b * NB + node) * KNN;
#pragma unroll
  for (int k = 0; k < KNN; ++k) {
    potbuf[base + k] = __expf(-bestd[k]);
    idxbuf[base + k] = idxs[tid][k];
  }
}

// edge messages pot*h[idx] -> mean/max over K, written into cat96 cols 0..63
__global__ void k_agg(const float* __restrict__ hbuf,
                      const float* __restrict__ potbuf,
                      const int* __restrict__ idxbuf,
                      float* __restrict__ cat96) {
  int lane = threadIdx.x & 31;
  int w = threadIdx.x >> 5;
  int row = blockIdx.x * 8 + w;  // 0..NTOT-1
  int b = row >> 13;             // row / NB
  const float* hb = hbuf + (size_t)b * NB * DD;
  size_t base = (size_t)row * KNN;
  float mean = 0.0f, mxv = -INFINITY;
  for (int k = 0; k < KNN; ++k) {
    float p = potbuf[base + k];
    int id = idxbuf[base + k];
    float hv = hb[(size_t)id * DD + lane];
    float m = p * hv;
    mean += m;
    mxv = fmaxf(mxv, m);
  }
  cat96[(size_t)row * 96 + lane] = mean * (1.0f / KNN);
  cat96[(size_t)row * 96 + 32 + lane] = mxv;
}

// xgn = [agg | x1] (96) @ lin_W + lin_b -> xc cols 0..31
__global__ void k_gn_lin(const float* __restrict__ cat96,
                         const float* __restrict__ W,
                         const float* __restrict__ bias,
                         float* __restrict__ xc) {
  int tid = threadIdx.x, lane = tid & 31, w = tid >> 5;
  int row0 = blockIdx.x * 128 + w * 16;
  v8f a0 = {}, a1 = {};
#pragma unroll
  for (int k0 = 0; k0 < 96; k0 += 4) {
    v2f av = load_a(cat96, 96, row0, k0, lane);
    a0 = wmma4(av, load_b(W, 0, k0, lane), a0);
    a1 = wmma4(av, load_b(W, 16, k0, lane), a1);
  }
  int n = lane & 15;
  float bb0 = bias[n], bb1 = bias[n + 16];
#pragma unroll
  for (int r = 0; r < 8; ++r) { a0[r] += bb0; a1[r] += bb1; }
  store_d(xc, 68, row0, 0, lane, a0);
  store_d(xc, 68, row0, 16, lane, a1);
}

// post_gravnet: ELU(ELU(xc@W1+b1)@W2+b2) -> out_raw, accumulate BN2 stats
__global__ void k_post_mlp(const float* __restrict__ xc,
                           const float* __restrict__ W1,
                           const float* __restrict__ b1,
                           const float* __restrict__ W2,
                           const float* __restrict__ b2,
                           float* __restrict__ out_raw,
                           float* __restrict__ bnstats) {
  __shared__ float w1p[68][32];  // post_W1 zero-padded K 67->68
  __shared__ float xp[8][16][32];
  __shared__ float bnacc[64];
  int tid = threadIdx.x;
  if (tid < 64) bnacc[tid] = 0.0f;
  for (int i = tid; i < 68 * 32; i += 256) {
    int kr = i >> 5, c = i & 31;
    w1p[kr][c] = (kr < 67) ? W1[kr * DD + c] : 0.0f;
  }
  __syncthreads();
  int lane = tid & 31, w = tid >> 5;
  int row0 = blockIdx.x * 128 + w * 16;
  int n = lane & 15;

  v8f a0 = {}, a1 = {};
#pragma unroll
  for (int k0 = 0; k0 < 68; k0 += 4) {  // K=67 padded (xc col 67 is zeroed)
    int kk = k0 + ((lane & 16) ? 2 : 0);
    v2f av = load_a(xc, 68, row0, k0, lane);
    v2f bv0; bv0.x = w1p[kk][n];      bv0.y = w1p[kk + 1][n];
    v2f bv1; bv1.x = w1p[kk][n + 16]; bv1.y = w1p[kk + 1][n + 16];
    a0 = wmma4(av, bv0, a0);
    a1 = wmma4(av, bv1, a1);
  }
  float bb0 = b1[n], bb1 = b1[n + 16];
#pragma unroll
  for (int r = 0; r < 8; ++r) { a0[r] += bb0; a1[r] += bb1; }
  a0 = elu8(a0); a1 = elu8(a1);

  int rb = (lane & 16) ? 8 : 0;
#pragma unroll
  for (int r = 0; r < 8; ++r) {
    xp[w][rb + r][n] = a0[r];
    xp[w][rb + r][n + 16] = a1[r];
  }
  __builtin_amdgcn_wave_barrier();

  v8f o0 = {}, o1 = {};
#pragma unroll
  for (int k0 = 0; k0 < 32; k0 += 4) {
    int kk = k0 + ((lane & 16) ? 2 : 0);
    v2f av; av.x = xp[w][n][kk]; av.y = xp[w][n][kk + 1];
    o0 = wmma4(av, load_b(W2, 0, k0, lane), o0);
    o1 = wmma4(av, load_b(W2, 16, k0, lane), o1);
  }
  bb0 = b2[n]; bb1 = b2[n + 16];
#pragma unroll
  for (int r = 0; r < 8; ++r) { o0[r] += bb0; o1[r] += bb1; }
  o0 = elu8(o0); o1 = elu8(o1);
  store_d(out_raw, DD, row0, 0, lane, o0);
  store_d(out_raw, DD, row0, 16, lane, o1);

  float s0 = 0, q0 = 0, s1 = 0, q1 = 0;
#pragma unroll
  for (int r = 0; r < 8; ++r) {
    s0 += o0[r]; q0 += o0[r] * o0[r];
    s1 += o1[r]; q1 += o1[r] * o1[r];
  }
  atomicAdd(&bnacc[n], s0);
  atomicAdd(&bnacc[n + 16], s1);
  atomicAdd(&bnacc[32 + n], q0);
  atomicAdd(&bnacc[32 + n + 16], q1);
  __syncthreads();
  if (tid < 64) atomicAdd(&bnstats[64 + tid], bnacc[tid]);
}

__global__ void k_bn2_out(const float* __restrict__ out_raw,
                          const float* __restrict__ bnstats,
                          const float* __restrict__ g,
                          const float* __restrict__ bvec,
                          float* __restrict__ out) {
  int i = blockIdx.x * 256 + threadIdx.x;  // NTOT*DD elements
  int ch = i & 31;
  float mu = bnstats[64 + ch] * (1.0f / NTOT);
  float var = bnstats[96 + ch] * (1.0f / NTOT) - mu * mu;
  float s = g[ch] * rsqrtf(var + BN_EPS);
  out[i] = out_raw[i] * s + (bvec[ch] - mu * s);
}

// ---------------------------------------------------------------- launch
extern "C" void kernel_launch(void* const* d_in, const int* in_sizes, int n_in,
                              void* d_out, int out_size, void* d_ws,
                              size_t ws_size, hipStream_t stream) {
  const float* x      = (const float*)d_in[0];
  const float* preW1  = (const float*)d_in[1];
  const float* preb1  = (const float*)d_in[2];
  const float* preW2  = (const float*)d_in[3];
  const float* preb2  = (const float*)d_in[4];
  const float* bn1g   = (const float*)d_in[5];
  const float* bn1b   = (const float*)d_in[6];
  const float* linsW  = (const float*)d_in[7];
  const float* linhW  = (const float*)d_in[8];
  const float* linhb  = (const float*)d_in[9];
  const float* linW   = (const float*)d_in[10];
  const float* linb   = (const float*)d_in[11];
  const float* postW1 = (const float*)d_in[12];
  const float* postb1 = (const float*)d_in[13];
  const float* postW2 = (const float*)d_in[14];
  const float* postb2 = (const float*)d_in[15];
  const float* bn2g   = (const float*)d_in[16];
  const float* bn2b   = (const float*)d_in[17];

  float* ws = (float*)d_ws;
  float* x1_raw  = ws;                          // NTOT*32
  float* hbuf    = x1_raw + (size_t)NTOT * 32;  // NTOT*32
  float* cat96   = hbuf + (size_t)NTOT * 32;    // NTOT*96
  float* xc      = cat96 + (size_t)NTOT * 96;   // NTOT*68
  float* sbuf    = xc + (size_t)NTOT * 68;      // NTOT*3
  float* potbuf  = sbuf + (size_t)NTOT * 3;     // NTOT*40
  int*   idxbuf  = (int*)(potbuf + (size_t)NTOT * 40);  // NTOT*40
  float* out_raw = (float*)(idxbuf + (size_t)NTOT * 40);// NTOT*32
  float* bnstats = out_raw + (size_t)NTOT * 32; // 128

  k_init<<<1, 128, 0, stream>>>(bnstats);
  k_pre_mlp<<<128, 256, 0, stream>>>(x, preW1, preb1, preW2, preb2, x1_raw, bnstats);
  k_bn1_proj<<<128, 256, 0, stream>>>(x1_raw, bnstats, bn1g, bn1b, linhW, linhb,
                                      linsW, hbuf, sbuf, cat96, xc);
  k_knn<<<dim3(NB / 128, 2), 128, 0, stream>>>(sbuf, potbuf, idxbuf);
  k_agg<<<NTOT / 8, 256, 0, stream>>>(hbuf, potbuf, idxbuf, cat96);
  k_gn_lin<<<128, 256, 0, stream>>>(cat96, linW, linb, xc);
  k_post_mlp<<<128, 256, 0, stream>>>(xc, postW1, postb1, postW2, postb2,
                                      out_raw, bnstats);
  k_bn2_out<<<(NTOT * DD) / 256, 256, 0, stream>>>(out_raw, bnstats, bn2g, bn2b,
                                                   (float*)d_out);
}